// PulseFloatingPointEncoder_76312978916069
// MI455X (gfx1250) — compile-verified
//
#include <hip/hip_runtime.h>

// PulseFloatingPointEncoder: element-wise re-encode of fp32 -> 8 indicator
// floats [sign, e3,e2,e1,e0, m0,m1,m2] (E4M3, bias 7, truncating mantissa,
// subnormals below 2^-6, no rounding).
//
// Closed-form equivalent of the reference 20-step scan:
//   u  = floor(|x| * 2^10)            (exact: pow2 scale + trunc convert)
//   hb = index of highest set bit of u
//   u >= 16  (normal):   e = (hb-3) & 0xF ; m = bits hb-1, hb-2, hb-3 of u
//   u <  16  (subnorm):  e = 0           ; m = bits 3, 2, 1 of u
// The t0=0 case (|x| in [512,1024)) gives e = 16 & 0xF = 0, exactly like the
// reference (counter=16 has no bits in [3:0]). |x| >= 1024 falls back to the
// literal greedy bit-extraction loop so any input stays bit-exact.

typedef float v4f __attribute__((ext_vector_type(4)));

__global__ __launch_bounds__(256) void pulse_encode_kernel(
    const float* __restrict__ x, float* __restrict__ out, int n) {
  int i = blockIdx.x * blockDim.x + threadIdx.x;
  if (i >= n) return;

  // Read-once input: non-temporal load (TH=NT) — don't pollute WGP$/L2.
  float xv = __builtin_nontemporal_load(x + i);

  float sgn = (xv < 0.0f) ? 1.0f : 0.0f;   // matches (xf < 0); -0.0 -> 0
  float a = fabsf(xv);

  unsigned u;
  if (a < 1024.0f) {
    // Exact: a*1024 is a pow2 scale (no rounding), u32 convert truncates.
    // Bit k of u == binary digit of |x| at weight 2^(k-10), k = 19..0.
    u = (unsigned)(a * 1024.0f);
  } else {
    // Rare/never path for N(0,1) data: replicate the reference's greedy scan.
    float V = a;
    unsigned ub = 0;
    float thr = 512.0f;
#pragma unroll
    for (int t = 0; t < 20; ++t) {
      if (V >= thr) { V -= thr; ub |= 1u << (19 - t); }
      thr *= 0.5f;
    }
    u = ub;
  }

  unsigned e, b0, b1, b2;
  if (u >= 16u) {
    // Normal: leading bit position hb>=4  <=>  first fire at t0 = 19-hb <= 15.
    unsigned hb = 31u - (unsigned)__builtin_clz(u);
    e  = (hb - 3u) & 0xFu;            // exponent field (wraps at hb=19 like ref)
    b0 = (u >> (hb - 1u)) & 1u;       // mantissa = 3 bits after the leading 1
    b1 = (u >> (hb - 2u)) & 1u;
    b2 = (u >> (hb - 3u)) & 1u;
  } else {
    // Subnormal (|x| < 2^-6): e = 0, mantissa = raw bits at 2^-7, 2^-8, 2^-9.
    e  = 0u;
    b0 = (u >> 3) & 1u;
    b1 = (u >> 2) & 1u;
    b2 = (u >> 1) & 1u;
  }

  // Pack [sign, e3,e2,e1] and [e0, m0,m1,m2]; e bits are emitted MSB-first
  // (reference reverses e_reg before concat).
  v4f lo = { sgn,
             (float)((e >> 3) & 1u),
             (float)((e >> 2) & 1u),
             (float)((e >> 1) & 1u) };
  v4f hi = { (float)(e & 1u),
             (float)b0,
             (float)b1,
             (float)b2 };

  // 512 MiB write-once output (> 192 MB L2): stream with non-temporal
  // global_store_b128 pairs; i*8 floats => 32-byte aligned.
  v4f* o = (v4f*)(out + (size_t)i * 8u);
  __builtin_nontemporal_store(lo, o);
  __builtin_nontemporal_store(hi, o + 1);
}

extern "C" void kernel_launch(void* const* d_in, const int* in_sizes, int n_in,
                              void* d_out, int out_size, void* d_ws, size_t ws_size,
                              hipStream_t stream) {
  const float* x = (const float*)d_in[0];
  float* out = (float*)d_out;
  int n = in_sizes[0];                 // 8*2048*1024 = 16,777,216
  const int block = 256;               // 8 wave32s per block
  const int grid = (n + block - 1) / block;
  pulse_encode_kernel<<<grid, block, 0, stream>>>(x, out, n);
}